// GMMLinear_74990128988171
// MI455X (gfx1250) — compile-verified
//
#include <hip/hip_runtime.h>
#include <math.h>

// GMM-imputed linear layer, fully fused, fp32 WMMA (v_wmma_f32_16x16x4_f32).
// B=8192, IN=1024, OUT=1024, C=8 (fixed by the reference).

typedef float v2f __attribute__((ext_vector_type(2)));
typedef float v8f __attribute__((ext_vector_type(8)));

#define NC   8
#define DIN  1024
#define DOUT 1024
#define DB   8192

__device__ __forceinline__ v8f wmma_f32(v2f a, v2f b, v8f c) {
  // 8-arg form: (neg_a, A, neg_b, B, c_mod, C, reuse_a, reuse_b)
  return __builtin_amdgcn_wmma_f32_16x16x4_f32(false, a, false, b, (short)0, c,
                                               false, false);
}

__global__ __launch_bounds__(128, 1) void gmm_linear_fused(
    const float* __restrict__ x, const float* __restrict__ W,
    const float* __restrict__ bias, const float* __restrict__ logw,
    const float* __restrict__ means, const float* __restrict__ covs,
    float* __restrict__ out) {
  // Per-component K-scales in LDS: 8*1024 floats each (32KB + 32KB of 320KB WGP LDS)
  __shared__ float s_mean[NC * DIN];
  __shared__ float s_cov[NC * DIN];
  for (int i = threadIdx.x; i < NC * DIN; i += 128) {
    s_mean[i] = means[i];
    s_cov[i]  = fabsf(covs[i]);
  }
  __syncthreads();

  const int lane = threadIdx.x & 31;
  const int wave = threadIdx.x >> 5;
  const int half = lane >> 4;   // 0: lanes 0-15, 1: lanes 16-31
  const int l15  = lane & 15;

  const int m0 = (blockIdx.x * 4 + wave) * 16;  // 16-row stripe per wave
  const int n0 = blockIdx.y * 16;               // 16-col stripe per block.y

  // A (16x4 f32): lane holds row M=l15, K = k0 + 2*half + {0,1}  -> b64 loads
  const float* xrow = x + (size_t)(m0 + l15) * DIN + 2 * half;
  // B (4x16 f32): VGPR j holds row K = k0 + 2*half + j, col N = n0+l15
  const float* wptr = W + (size_t)(2 * half) * DOUT + (n0 + l15);

  v8f acc_base = {};
  v8f accY[NC];
  v8f accV[NC];
#pragma unroll
  for (int i = 0; i < NC; ++i) {
    accY[i] = (v8f){};
    accV[i] = (v8f){};
  }

  float masksum = 0.0f;  // per-lane partial count of missing entries in row m0+l15

  for (int k0 = 0; k0 < DIN; k0 += 4) {
    float xv0 = xrow[k0];
    float xv1 = xrow[k0 + 1];
    float wv0 = wptr[(size_t)k0 * DOUT];
    float wv1 = wptr[(size_t)(k0 + 1) * DOUT];

    bool nan0 = (xv0 != xv0);
    bool nan1 = (xv1 != xv1);
    float mf0 = nan0 ? 1.0f : 0.0f;
    float mf1 = nan1 ? 1.0f : 0.0f;
    float xc0 = nan0 ? 0.0f : xv0;
    float xc1 = nan1 ? 0.0f : xv1;
    masksum += mf0 + mf1;

    v2f aClean = {xc0, xc1};
    v2f bW  = {wv0, wv1};
    v2f bW2 = {wv0 * wv0, wv1 * wv1};  // W^2 tile on the fly (no extra pass)

    acc_base = wmma_f32(aClean, bW, acc_base);  // base = x_clean @ W

    const int kbase = k0 + 2 * half;
#pragma unroll
    for (int i = 0; i < NC; ++i) {
      const int ki = i * DIN + kbase;
      // (maskf * mean_i) @ W : diag scale applied on the A side
      v2f am = {mf0 * s_mean[ki], mf1 * s_mean[ki + 1]};
      accY[i] = wmma_f32(am, bW, accY[i]);
      // (maskf * |cov_i|) @ W^2
      v2f ac = {mf0 * s_cov[ki], mf1 * s_cov[ki + 1]};
      accV[i] = wmma_f32(ac, bW2, accV[i]);
    }
  }

  // Row-missing flag: lane l and l^16 cover complementary K residues of row l&15;
  // xor-combine gives every lane the full row count for row (lane&15).
  float rowmask = masksum + __shfl_xor(masksum, 16, 32);

  // softmax over the 8 log-weights (tiny, L2-resident)
  float pw[NC];
  float pmax = -INFINITY;
#pragma unroll
  for (int i = 0; i < NC; ++i) {
    pw[i] = logw[i];
    pmax = fmaxf(pmax, pw[i]);
  }
  float psum = 0.0f;
#pragma unroll
  for (int i = 0; i < NC; ++i) {
    pw[i] = __expf(pw[i] - pmax);
    psum += pw[i];
  }
  const float pinv = 1.0f / psum;

  const float bv = bias[n0 + l15];
  const float INV_SQRT_2PI = 0.3989422804014327f;
  const float INV_SQRT_2   = 0.7071067811865476f;

  // C/D layout: VGPR v, lanes 0-15 -> M=v, lanes 16-31 -> M=8+v; N=lane&15
#pragma unroll
  for (int v = 0; v < 8; ++v) {
    const int rloc = 8 * half + v;
    const float rm = __shfl(rowmask, rloc, 32);
    const float basev = acc_base[v] + bv;
    float result;
    if (rm > 0.0f) {
      float s = 0.0f;
#pragma unroll
      for (int i = 0; i < NC; ++i) {
        float a   = basev + accY[i][v];
        float var = accV[i][v];
        float denom = sqrtf(var > 0.0f ? var : 1.0f);
        float z = a / denom;
        float nrv = __expf(-0.5f * z * z) * INV_SQRT_2PI +
                    0.5f * z * (1.0f + erff(z * INV_SQRT_2));
        s += pw[i] * pinv * nrv;
      }
      result = s;
    } else {
      result = fmaxf(basev, 0.0f);
    }
    out[(size_t)(m0 + rloc) * DOUT + (n0 + l15)] = result;
  }
}

extern "C" void kernel_launch(void* const* d_in, const int* in_sizes, int n_in,
                              void* d_out, int out_size, void* d_ws,
                              size_t ws_size, hipStream_t stream) {
  (void)in_sizes; (void)n_in; (void)out_size; (void)d_ws; (void)ws_size;
  const float* x     = (const float*)d_in[0];  // [8192,1024]
  const float* W     = (const float*)d_in[1];  // [1024,1024]
  const float* b     = (const float*)d_in[2];  // [1024]
  const float* logw  = (const float*)d_in[3];  // [8]
  const float* means = (const float*)d_in[4];  // [8,1024]
  const float* covs  = (const float*)d_in[5];  // [8,1024]
  float* out = (float*)d_out;                  // [8192,1024]

  dim3 grid(DB / (16 * 4), DOUT / 16);  // 128 x 64 blocks
  dim3 block(128);                      // 4 waves, one 16x16 tile-set each
  gmm_linear_fused<<<grid, block, 0, stream>>>(x, W, b, logw, means, covs, out);
}